// PrepareEnvGCN_59236188947119
// MI455X (gfx1250) — compile-verified
//
#include <hip/hip_runtime.h>
#include <hip/hip_bf16.h>

// ---------------------------------------------------------------------------
// TransformerConv x4 GNN for MI455X (gfx1250, wave32).
// Node linear transforms (x@Wq/Wk/Wv/Ws) run on V_WMMA_F32_16X16X4_F32:
// one wave handles a 16-node tile, K=16 accumulated over 4 WMMA steps,
// 4 weight matrices => 16 v_wmma per wave. Edge softmax/scatter passes use
// templated OUT_C with float4 (global_load_b128) gathers and native f32
// global atomics (add / ordered-uint max). Whole working set (~125 MB)
// fits in the 192 MB L2, so minimizing VMEM instruction count is the lever.
// ---------------------------------------------------------------------------

typedef __attribute__((ext_vector_type(2))) float v2f;
typedef __attribute__((ext_vector_type(8))) float v8f;

#define NNODES 100000
#define NEDGES 3200000
#define NGRAPH 64

// ---------------- WMMA node transform: q/k/v/skip = hin @ W + b ------------
__global__ __launch_bounds__(256) void node_transform_wmma(
    const float* __restrict__ hin,                       // [N,16] zero-padded
    const float* __restrict__ Wq, const float* __restrict__ bq,
    const float* __restrict__ Wk, const float* __restrict__ bk,
    const float* __restrict__ Wv, const float* __restrict__ bv,
    const float* __restrict__ Ws, const float* __restrict__ bs,
    float* __restrict__ qo, float* __restrict__ ko,
    float* __restrict__ vo, float* __restrict__ so,
    int n_nodes, int in_c, int out_c)
{
    __shared__ float sW[4][16][16];   // [which][k][n], zero-padded to 16x16
    __shared__ float sB[4][16];
    const int t = threadIdx.x;

    // cooperative zero-padded weight staging
    for (int i = t; i < 4 * 256; i += 256) {
        int w  = i >> 8;
        int kk = (i >> 4) & 15;
        int nn = i & 15;
        const float* W = (w == 0) ? Wq : (w == 1) ? Wk : (w == 2) ? Wv : Ws;
        sW[w][kk][nn] = (kk < in_c && nn < out_c) ? W[kk * out_c + nn] : 0.0f;
    }
    if (t < 64) {
        int w = t >> 4, nn = t & 15;
        const float* B = (w == 0) ? bq : (w == 1) ? bk : (w == 2) ? bv : bs;
        sB[w][nn] = (nn < out_c) ? B[nn] : 0.0f;
    }
    __syncthreads();

    const int wave = t >> 5;
    const int lane = t & 31;
    const int tile = blockIdx.x * 8 + wave;
    if (tile * 16 >= n_nodes) return;          // wave-uniform: EXEC stays all-1s

    const int row  = lane & 15;                // A-matrix M index
    const int khi  = (lane >> 4) * 2;          // lanes 16-31 hold K+2,K+3
    const int node = tile * 16 + row;
    const int ncol = lane & 15;                // B/C/D column (N)
    const int mhi  = (lane >> 4) * 8;          // C/D rows 8..15 in lanes 16-31

    // A fragments for the 4 K-steps (16x4 f32: 2 VGPRs/lane)
    v2f a[4];
#pragma unroll
    for (int ks = 0; ks < 4; ++ks) {
        int k0 = ks * 4 + khi;
        a[ks].x = hin[node * 16 + k0];
        a[ks].y = hin[node * 16 + k0 + 1];
    }

#pragma unroll
    for (int w = 0; w < 4; ++w) {
        v8f c = {};
#pragma unroll
        for (int ks = 0; ks < 4; ++ks) {
            int k0 = ks * 4 + khi;
            v2f b;
            b.x = sW[w][k0][ncol];
            b.y = sW[w][k0 + 1][ncol];
            c = __builtin_amdgcn_wmma_f32_16x16x4_f32(
                    false, a[ks], false, b, (short)0, c, false, false);
        }
        const float bias = sB[w][ncol];
        float* dstp = (w == 0) ? qo : (w == 1) ? ko : (w == 2) ? vo : so;
#pragma unroll
        for (int r = 0; r < 8; ++r) {
            int m = r + mhi;
            dstp[(tile * 16 + m) * 16 + ncol] = c[r] + bias;
        }
    }
}

// ---------------- per-layer scratch clear ----------------------------------
__global__ void clear_layer_kernel(unsigned* __restrict__ amax_u,
                                   float* __restrict__ denom,
                                   float* __restrict__ acc, int n_nodes)
{
    int i = blockIdx.x * blockDim.x + threadIdx.x;
    if (i < n_nodes) { amax_u[i] = 0u; denom[i] = 0.0f; }
    if (i < n_nodes * 16) acc[i] = 0.0f;
}

// ---------------- edge pass A: logits + segment max ------------------------
template <int OUT_C>
__global__ void edge_logits_kernel(
    const int* __restrict__ src, const int* __restrict__ dst,
    const float* __restrict__ eattr,
    const float* __restrict__ q, const float* __restrict__ k,
    const float* __restrict__ We,
    float* __restrict__ alpha, unsigned* __restrict__ amax_u,
    float inv_sqrt, int n_edges)
{
    int e = blockIdx.x * blockDim.x + threadIdx.x;
    if (e >= n_edges) return;
    const int s = src[e], d = dst[e];
    const float ea = eattr[e];

    float we[OUT_C];
#pragma unroll
    for (int c = 0; c < OUT_C; ++c) we[c] = We[c];   // uniform -> s_load

    const float4* qd = reinterpret_cast<const float4*>(q + (size_t)d * 16);
    const float4* ks = reinterpret_cast<const float4*>(k + (size_t)s * 16);

    float acc = 0.0f;
#pragma unroll
    for (int c4 = 0; c4 < OUT_C / 4; ++c4) {
        const float4 qv = qd[c4];                    // global_load_b128
        const float4 kv = ks[c4];
        acc += qv.x * (kv.x + ea * we[c4 * 4 + 0]);
        acc += qv.y * (kv.y + ea * we[c4 * 4 + 1]);
        acc += qv.z * (kv.z + ea * we[c4 * 4 + 2]);
        acc += qv.w * (kv.w + ea * we[c4 * 4 + 3]);
    }
    acc *= inv_sqrt;
    alpha[e] = acc;
    unsigned u = __float_as_uint(acc);
    u = (u & 0x80000000u) ? ~u : (u | 0x80000000u);  // order-preserving map
    atomicMax(&amax_u[d], u);
}

// ---------------- edge pass B: exp + weighted scatter ----------------------
template <int OUT_C>
__global__ void edge_accum_kernel(
    const int* __restrict__ src, const int* __restrict__ dst,
    const float* __restrict__ eattr,
    const float* __restrict__ v, const float* __restrict__ We,
    const float* __restrict__ alpha, const unsigned* __restrict__ amax_u,
    float* __restrict__ denom, float* __restrict__ acc,
    int n_edges)
{
    int e = blockIdx.x * blockDim.x + threadIdx.x;
    if (e >= n_edges) return;
    const int s = src[e], d = dst[e];
    const unsigned u = amax_u[d];
    const float am = (u & 0x80000000u) ? __uint_as_float(u ^ 0x80000000u)
                                       : __uint_as_float(~u);
    const float ex = expf(alpha[e] - am);
    atomicAdd(&denom[d], ex);
    const float ea = eattr[e];

    float we[OUT_C];
#pragma unroll
    for (int c = 0; c < OUT_C; ++c) we[c] = We[c];

    const float4* vs = reinterpret_cast<const float4*>(v + (size_t)s * 16);
    float* accd = acc + (size_t)d * 16;
#pragma unroll
    for (int c4 = 0; c4 < OUT_C / 4; ++c4) {
        const float4 vv = vs[c4];                    // global_load_b128
        atomicAdd(&accd[c4 * 4 + 0], ex * (vv.x + ea * we[c4 * 4 + 0]));
        atomicAdd(&accd[c4 * 4 + 1], ex * (vv.y + ea * we[c4 * 4 + 1]));
        atomicAdd(&accd[c4 * 4 + 2], ex * (vv.z + ea * we[c4 * 4 + 2]));
        atomicAdd(&accd[c4 * 4 + 3], ex * (vv.w + ea * we[c4 * 4 + 3]));
    }
}

// ---------------- node finalize: normalize + skip + leaky ------------------
template <int OUT_C, bool LEAKY>
__global__ void node_finalize_kernel(
    const float* __restrict__ acc, const float* __restrict__ denom,
    const float* __restrict__ skip, float* __restrict__ hout,
    int n_nodes)
{
    int node = blockIdx.x * blockDim.x + threadIdx.x;
    if (node >= n_nodes) return;
    const float den = denom[node];
    const float inv = (den > 0.0f) ? 1.0f / den : 0.0f;

    const float4* a4 = reinterpret_cast<const float4*>(acc  + (size_t)node * 16);
    const float4* s4 = reinterpret_cast<const float4*>(skip + (size_t)node * 16);
    float4*       o4 = reinterpret_cast<float4*>(hout + (size_t)node * 16);

    float row[16];
#pragma unroll
    for (int c = 0; c < 16; ++c) row[c] = 0.0f;

#pragma unroll
    for (int c4 = 0; c4 < OUT_C / 4; ++c4) {
        const float4 av = a4[c4];
        const float4 sv = s4[c4];
        float r0 = av.x * inv + sv.x;
        float r1 = av.y * inv + sv.y;
        float r2 = av.z * inv + sv.z;
        float r3 = av.w * inv + sv.w;
        if (LEAKY) {
            r0 = (r0 > 0.0f) ? r0 : 0.01f * r0;
            r1 = (r1 > 0.0f) ? r1 : 0.01f * r1;
            r2 = (r2 > 0.0f) ? r2 : 0.01f * r2;
            r3 = (r3 > 0.0f) ? r3 : 0.01f * r3;
        }
        row[c4 * 4 + 0] = r0; row[c4 * 4 + 1] = r1;
        row[c4 * 4 + 2] = r2; row[c4 * 4 + 3] = r3;
    }
#pragma unroll
    for (int c4 = 0; c4 < 4; ++c4)
        o4[c4] = make_float4(row[c4 * 4], row[c4 * 4 + 1],
                             row[c4 * 4 + 2], row[c4 * 4 + 3]);  // b128 stores
}

// ---------------- pooling + classifier -------------------------------------
__global__ void pool_clear_kernel(float* __restrict__ pool)
{
    int i = threadIdx.x;
    if (i < NGRAPH * 5) pool[i] = 0.0f;
}

__global__ void pool_accum_kernel(const float* __restrict__ h,
                                  const int* __restrict__ batch,
                                  float* __restrict__ pool, int n_nodes)
{
    int i = blockIdx.x * blockDim.x + threadIdx.x;
    if (i >= n_nodes) return;
    const int g = batch[i];
    const float4 hv = *reinterpret_cast<const float4*>(h + (size_t)i * 16);
    atomicAdd(&pool[g * 5 + 0], hv.x);
    atomicAdd(&pool[g * 5 + 1], hv.y);
    atomicAdd(&pool[g * 5 + 2], hv.z);
    atomicAdd(&pool[g * 5 + 3], hv.w);
    atomicAdd(&pool[g * 5 + 4], 1.0f);
}

__global__ void classify_kernel(const float* __restrict__ pool,
                                const float* __restrict__ cw,
                                const float* __restrict__ cb,
                                float* __restrict__ out)
{
    int g = threadIdx.x;
    if (g >= NGRAPH) return;
    const float cnt = fmaxf(pool[g * 5 + 4], 1.0f);
    float r = cb[0];
    for (int c = 0; c < 4; ++c)
        r += (pool[g * 5 + c] / cnt) * cw[c] + pool[g * 5 + c] * cw[4 + c];
    out[g] = r;
}

// ---------------------------------------------------------------------------
extern "C" void kernel_launch(void* const* d_in, const int* in_sizes, int n_in,
                              void* d_out, int out_size, void* d_ws, size_t ws_size,
                              hipStream_t stream)
{
    // ---- resolve big inputs by their unique flat sizes ----
    int ix = -1, iei = -1, iea = -1, ib = -1;
    for (int i = 0; i < n_in; ++i) {
        if      (in_sizes[i] == NNODES * 16) ix  = i;   // x: 1,600,000
        else if (in_sizes[i] == 2 * NEDGES)  iei = i;   // edge_index: 6,400,000
        else if (in_sizes[i] == NEDGES)      iea = i;   // edge_attr: 3,200,000
        else if (in_sizes[i] == NNODES)      ib  = i;   // batch: 100,000
    }
    int rem[64]; int nr = 0;
    for (int i = 0; i < n_in && nr < 64; ++i)
        if (i != ix && i != iei && i != iea && i != ib) rem[nr++] = i;

    // param leaf pointers
    const float *Wq[4], *bq[4], *Wk[4], *bk[4], *Wv[4], *bv[4],
                *We[4], *Ws[4], *bs[4], *clsw = nullptr, *clsb = nullptr;
    if (nr >= 38 && in_sizes[rem[0]] == 1) {
        // jax-sorted order: cls_b, cls_w, then per layer {We,Wk,Wq,Ws,Wv,bk,bq,bs,bv}
        clsb = (const float*)d_in[rem[0]];
        clsw = (const float*)d_in[rem[1]];
        for (int L = 0; L < 4; ++L) {
            int o = 2 + L * 9;
            We[L] = (const float*)d_in[rem[o + 0]];
            Wk[L] = (const float*)d_in[rem[o + 1]];
            Wq[L] = (const float*)d_in[rem[o + 2]];
            Ws[L] = (const float*)d_in[rem[o + 3]];
            Wv[L] = (const float*)d_in[rem[o + 4]];
            bk[L] = (const float*)d_in[rem[o + 5]];
            bq[L] = (const float*)d_in[rem[o + 6]];
            bs[L] = (const float*)d_in[rem[o + 7]];
            bv[L] = (const float*)d_in[rem[o + 8]];
        }
    } else {
        // insertion order: per layer {Wq,bq,Wk,bk,Wv,bv,We,Ws,bs}, then cls_w, cls_b
        for (int L = 0; L < 4; ++L) {
            int o = L * 9;
            Wq[L] = (const float*)d_in[rem[o + 0]];
            bq[L] = (const float*)d_in[rem[o + 1]];
            Wk[L] = (const float*)d_in[rem[o + 2]];
            bk[L] = (const float*)d_in[rem[o + 3]];
            Wv[L] = (const float*)d_in[rem[o + 4]];
            bv[L] = (const float*)d_in[rem[o + 5]];
            We[L] = (const float*)d_in[rem[o + 6]];
            Ws[L] = (const float*)d_in[rem[o + 7]];
            bs[L] = (const float*)d_in[rem[o + 8]];
        }
        clsw = (const float*)d_in[rem[36]];
        clsb = (const float*)d_in[rem[37]];
    }

    const float* x     = (const float*)d_in[ix];
    const int*   src   = (const int*)d_in[iei];            // edge_index row 0
    const int*   dstp  = ((const int*)d_in[iei]) + NEDGES; // row 1
    const float* eattr = (const float*)d_in[iea];
    const int*   batch = (const int*)d_in[ib];

    // ---- workspace carve-out ----
    char* w = (char*)d_ws;
    size_t off = 0;
    auto carve = [&](size_t bytes) -> char* {
        char* p = w + off;
        off += (bytes + 255) & ~(size_t)255;
        return p;
    };
    const size_t nf16 = (size_t)NNODES * 16 * sizeof(float);
    float*    h0     = (float*)carve(nf16);
    float*    h1     = (float*)carve(nf16);
    float*    qb     = (float*)carve(nf16);
    float*    kb     = (float*)carve(nf16);
    float*    vb     = (float*)carve(nf16);
    float*    sb     = (float*)carve(nf16);
    float*    accb   = (float*)carve(nf16);
    float*    denom  = (float*)carve((size_t)NNODES * sizeof(float));
    unsigned* amaxu  = (unsigned*)carve((size_t)NNODES * sizeof(unsigned));
    float*    alpha  = (float*)carve((size_t)NEDGES * sizeof(float));
    float*    pool   = (float*)carve((size_t)NGRAPH * 5 * sizeof(float));
    (void)ws_size;

    const int dims[4][2] = { {16, 16}, {16, 8}, {8, 8}, {8, 4} };

    const int tileBlocks = ((NNODES / 16) + 7) / 8;          // 8 waves/block
    const int nodeBlocks = (NNODES * 16 + 255) / 256;
    const int nBlocks    = (NNODES + 255) / 256;
    const int edgeBlocks = (NEDGES + 255) / 256;

    const float* hin = x;
    for (int L = 0; L < 4; ++L) {
        const int ic = dims[L][0], oc = dims[L][1];
        const float inv_sqrt = 1.0f / sqrtf((float)oc);
        float* hout = (L & 1) ? h1 : h0;
        const bool leaky = (L < 3);

        node_transform_wmma<<<tileBlocks, 256, 0, stream>>>(
            hin, Wq[L], bq[L], Wk[L], bk[L], Wv[L], bv[L], Ws[L], bs[L],
            qb, kb, vb, sb, NNODES, ic, oc);

        clear_layer_kernel<<<nodeBlocks, 256, 0, stream>>>(amaxu, denom, accb, NNODES);

        if (oc == 16) {
            edge_logits_kernel<16><<<edgeBlocks, 256, 0, stream>>>(
                src, dstp, eattr, qb, kb, We[L], alpha, amaxu, inv_sqrt, NEDGES);
            edge_accum_kernel<16><<<edgeBlocks, 256, 0, stream>>>(
                src, dstp, eattr, vb, We[L], alpha, amaxu, denom, accb, NEDGES);
            if (leaky) node_finalize_kernel<16, true ><<<nBlocks, 256, 0, stream>>>(accb, denom, sb, hout, NNODES);
            else       node_finalize_kernel<16, false><<<nBlocks, 256, 0, stream>>>(accb, denom, sb, hout, NNODES);
        } else if (oc == 8) {
            edge_logits_kernel<8><<<edgeBlocks, 256, 0, stream>>>(
                src, dstp, eattr, qb, kb, We[L], alpha, amaxu, inv_sqrt, NEDGES);
            edge_accum_kernel<8><<<edgeBlocks, 256, 0, stream>>>(
                src, dstp, eattr, vb, We[L], alpha, amaxu, denom, accb, NEDGES);
            if (leaky) node_finalize_kernel<8, true ><<<nBlocks, 256, 0, stream>>>(accb, denom, sb, hout, NNODES);
            else       node_finalize_kernel<8, false><<<nBlocks, 256, 0, stream>>>(accb, denom, sb, hout, NNODES);
        } else {
            edge_logits_kernel<4><<<edgeBlocks, 256, 0, stream>>>(
                src, dstp, eattr, qb, kb, We[L], alpha, amaxu, inv_sqrt, NEDGES);
            edge_accum_kernel<4><<<edgeBlocks, 256, 0, stream>>>(
                src, dstp, eattr, vb, We[L], alpha, amaxu, denom, accb, NEDGES);
            if (leaky) node_finalize_kernel<4, true ><<<nBlocks, 256, 0, stream>>>(accb, denom, sb, hout, NNODES);
            else       node_finalize_kernel<4, false><<<nBlocks, 256, 0, stream>>>(accb, denom, sb, hout, NNODES);
        }
        hin = hout;
    }

    pool_clear_kernel<<<1, 320, 0, stream>>>(pool);
    pool_accum_kernel<<<(NNODES + 255) / 256, 256, 0, stream>>>(hin, batch, pool, NNODES);
    classify_kernel<<<1, 64, 0, stream>>>(pool, clsw, clsb, (float*)d_out);
    (void)out_size;
}